// SAPCABlockV3_39676907888330
// MI455X (gfx1250) — compile-verified
//
#include <hip/hip_runtime.h>
#include <math.h>

typedef float v2f __attribute__((ext_vector_type(2)));
typedef float v8f __attribute__((ext_vector_type(8)));

#define C_DIM 256
#define HW    4096
#define BATCH 32
#define KSEL  64

// f32 WMMA 16x16x4 fragment helper (layout per CDNA5 ISA 7.12.2):
//   A 16x4 : lanes0-15 hold K={0,1}, lanes16-31 hold K={2,3}, M=lane&15
//   B 4x16 : mirrored split, N=lane&15
//   C/D    : VGPR v -> M = v + 8*(lane>=16), N = lane&15
static __device__ __forceinline__ v8f wmma_f32(v2f a, v2f b, v8f c) {
    return __builtin_amdgcn_wmma_f32_16x16x4_f32(
        false, a, false, b, (short)0, c, false, false);
}

// ---------------------------------------------------------------------------
// GEMM1: gx[b] = w1 @ x[b] + b1. Register-blocked: one wave owns a 64x64
// output tile (16 accumulators, 128 VGPRs); 8 waves/block tile N.
// Per k-step: 4 A-frag + 4 B-frag loads feed 16 WMMAs.
// ---------------------------------------------------------------------------
__global__ __launch_bounds__(256) void gemm_gx_kernel(
    const float* __restrict__ x, const float* __restrict__ w1,
    const float* __restrict__ b1, float* __restrict__ gxf)
{
    const int wave = threadIdx.x >> 5;
    const int lane = threadIdx.x & 31;
    const int half = lane >> 4;
    const int l    = lane & 15;
    const int b     = blockIdx.z;
    const int mBase = blockIdx.y * 64;
    const int nBase = (blockIdx.x * 8 + wave) * 64;

    const float* xb = x   + (size_t)b * C_DIM * HW;
    float*       gb = gxf + (size_t)b * C_DIM * HW;

    v8f acc[4][4] = {};
    for (int k0 = 0; k0 < C_DIM; k0 += 4) {
        const int ka = k0 + 2 * half;
        v2f af[4], bf[4];
#pragma unroll
        for (int i = 0; i < 4; ++i) {
            const float* pa = w1 + (size_t)(mBase + 16 * i + l) * C_DIM + ka;
            af[i].x = pa[0];
            af[i].y = pa[1];
        }
#pragma unroll
        for (int j = 0; j < 4; ++j) {
            const float* pb = xb + (size_t)ka * HW + nBase + 16 * j + l;
            bf[j].x = pb[0];
            bf[j].y = pb[HW];
        }
        if (k0 + 4 < C_DIM) {  // prefetch next K-slab of streaming operand
            __builtin_prefetch(xb + (size_t)(ka + 4) * HW + nBase + l, 0, 0);
        }
#pragma unroll
        for (int i = 0; i < 4; ++i)
#pragma unroll
            for (int j = 0; j < 4; ++j)
                acc[i][j] = wmma_f32(af[i], bf[j], acc[i][j]);
    }
#pragma unroll
    for (int i = 0; i < 4; ++i) {
#pragma unroll
        for (int v = 0; v < 8; ++v) {
            const int row = mBase + 16 * i + v + 8 * half;
            const float bias = b1[row];
#pragma unroll
            for (int j = 0; j < 4; ++j)
                gb[(size_t)row * HW + nBase + 16 * j + l] = acc[i][j][v] + bias;
        }
    }
}

// ---------------------------------------------------------------------------
// Subtract per-(b,c) spatial mean in place. One block per row of 4096.
// ---------------------------------------------------------------------------
__global__ __launch_bounds__(256) void mean_sub_kernel(float* __restrict__ gxf)
{
    __shared__ float red[256];
    const size_t base = (size_t)blockIdx.x * HW;
    float s = 0.f;
    for (int i = threadIdx.x; i < HW; i += 256) s += gxf[base + i];
    red[threadIdx.x] = s;
    __syncthreads();
    for (int off = 128; off > 0; off >>= 1) {
        if ((int)threadIdx.x < off) red[threadIdx.x] += red[threadIdx.x + off];
        __syncthreads();
    }
    const float mean = red[0] * (1.0f / HW);
    for (int i = threadIdx.x; i < HW; i += 256) gxf[base + i] -= mean;
}

// ---------------------------------------------------------------------------
// Covariance: cmat[b] = gxf[b] @ gxf[b]^T / BATCH.
// Block computes one 64x64 tile; 8 waves split K=4096 (512 each), each wave
// register-blocked 64x64, then a deterministic serialized LDS reduction.
// ---------------------------------------------------------------------------
__global__ __launch_bounds__(256) void gemm_cov_kernel(
    const float* __restrict__ gxf, float* __restrict__ cmat)
{
    const int wave = threadIdx.x >> 5;
    const int lane = threadIdx.x & 31;
    const int half = lane >> 4;
    const int l    = lane & 15;
    const int b     = blockIdx.z;
    const int mBase = blockIdx.y * 64;
    const int nBase = blockIdx.x * 64;

    const float* gb = gxf + (size_t)b * C_DIM * HW;
    __shared__ float csum[64 * 64];

    v8f acc[4][4] = {};
    const int kBeg = wave * (HW / 8);
    const int kEnd = kBeg + (HW / 8);
    for (int k0 = kBeg; k0 < kEnd; k0 += 4) {
        const int ka = k0 + 2 * half;
        v2f af[4], bf[4];
#pragma unroll
        for (int i = 0; i < 4; ++i) {
            const float* pa = gb + (size_t)(mBase + 16 * i + l) * HW + ka;
            af[i].x = pa[0];
            af[i].y = pa[1];
            const float* pb = gb + (size_t)(nBase + 16 * i + l) * HW + ka;  // B[k][n]=gxf[n][k]
            bf[i].x = pb[0];
            bf[i].y = pb[1];
        }
#pragma unroll
        for (int i = 0; i < 4; ++i)
#pragma unroll
            for (int j = 0; j < 4; ++j)
                acc[i][j] = wmma_f32(af[i], bf[j], acc[i][j]);
    }
    // deterministic cross-wave reduction (fixed order, no float atomics)
    for (int w = 0; w < 8; ++w) {
        if (wave == w) {
#pragma unroll
            for (int i = 0; i < 4; ++i)
#pragma unroll
                for (int v = 0; v < 8; ++v) {
                    const int r = 16 * i + v + 8 * half;
#pragma unroll
                    for (int j = 0; j < 4; ++j) {
                        const int idx = r * 64 + 16 * j + l;
                        if (w == 0) csum[idx] = acc[i][j][v];
                        else        csum[idx] += acc[i][j][v];
                    }
                }
        }
        __syncthreads();
    }
    float* cb = cmat + (size_t)b * C_DIM * C_DIM;
    for (int t = threadIdx.x; t < 64 * 64; t += 256) {
        const int r = t >> 6, c = t & 63;
        cb[(size_t)(mBase + r) * C_DIM + nBase + c] = csum[t] * (1.0f / BATCH);
    }
}

// ---------------------------------------------------------------------------
// Batched parallel cyclic Jacobi eigensolver. One WG (1024 thr) per batch.
// A (256x256, stride 257 -> conflict-free on 64 banks) in CDNA5 320KB LDS;
// eigvecs V in global ws (L2-resident: 8MB total). Ends by contracting
// top-64 eigvecs with w2 -> T_b = w2 @ W_b^T (C x C).
// ---------------------------------------------------------------------------
__global__ __launch_bounds__(1024) void jacobi_kernel(
    const float* __restrict__ cmat, const float* __restrict__ w2,
    float* __restrict__ V, float* __restrict__ T)
{
    __shared__ float As[C_DIM * 257];
    __shared__ float cs_c[128], cs_s[128];
    __shared__ int   pp[128], qq[128];
    __shared__ int   perm[C_DIM];

    const int    tid   = threadIdx.x;
    const int    b     = blockIdx.x;
    const size_t cbase = (size_t)b * C_DIM * C_DIM;
    float* Vb = V + cbase;
    float* Tb = T + cbase;

    for (int idx = tid; idx < C_DIM * C_DIM; idx += 1024) {
        As[(idx >> 8) * 257 + (idx & 255)] = cmat[cbase + idx];
        Vb[idx] = ((idx >> 8) == (idx & 255)) ? 1.0f : 0.0f;
    }
    __syncthreads();

    for (int sweep = 0; sweep < 8; ++sweep) {
        for (int r = 0; r < 255; ++r) {
            // 128 independent rotations: round-robin tournament pairing
            if (tid < 128) {
                const int k = tid;
                const int p = (k == 0) ? 0 : (1 + (k - 1 + r) % 255);
                const int q = 1 + (254 - k + r) % 255;
                const float app = As[p * 257 + p];
                const float aqq = As[q * 257 + q];
                const float apq = As[p * 257 + q];
                float c = 1.0f, s = 0.0f;
                if (fabsf(apq) > 1e-12f) {
                    const float tau = (aqq - app) / (2.0f * apq);
                    const float t =
                        copysignf(1.0f, tau) / (fabsf(tau) + sqrtf(1.0f + tau * tau));
                    c = 1.0f / sqrtf(1.0f + t * t);
                    s = t * c;
                }
                pp[k] = p; qq[k] = q; cs_c[k] = c; cs_s[k] = s;
            }
            __syncthreads();
            // phase 1: B = J^T A  (row pairs disjoint across k; lanes walk j
            // contiguously -> conflict-free)
            for (int idx = tid; idx < 128 * C_DIM; idx += 1024) {
                const int k = idx >> 8, j = idx & 255;
                const int p = pp[k], q = qq[k];
                const float c = cs_c[k], s = cs_s[k];
                const float ap = As[p * 257 + j], aq = As[q * 257 + j];
                As[p * 257 + j] = c * ap - s * aq;
                As[q * 257 + j] = s * ap + c * aq;
            }
            __syncthreads();
            // phase 2: A = B J, V = V J (column pairs disjoint; stride 257
            // over i -> each lane a distinct bank)
            for (int idx = tid; idx < 128 * C_DIM; idx += 1024) {
                const int k = idx >> 8, i = idx & 255;
                const int p = pp[k], q = qq[k];
                const float c = cs_c[k], s = cs_s[k];
                const float ip = As[i * 257 + p], iq = As[i * 257 + q];
                As[i * 257 + p] = c * ip - s * iq;
                As[i * 257 + q] = s * ip + c * iq;
                const float vp = Vb[i * C_DIM + p], vq = Vb[i * C_DIM + q];
                Vb[i * C_DIM + p] = c * vp - s * vq;
                Vb[i * C_DIM + q] = s * vp + c * vq;
            }
            __syncthreads();
        }
    }

    // rank-sort eigenvalues ascending (matches eigh ordering)
    if (tid < C_DIM) {
        const float vi = As[tid * 257 + tid];
        int rank = 0;
        for (int j = 0; j < C_DIM; ++j) {
            const float vj = As[j * 257 + j];
            if (vj < vi || (vj == vi && j < tid)) ++rank;
        }
        perm[rank] = tid;
    }
    __syncthreads();

    // T[o][c] = sum_kk w2[o][kk] * V[c][perm[192+kk]]
    for (int idx = tid; idx < C_DIM * C_DIM; idx += 1024) {
        const int o = idx >> 8, cc = idx & 255;
        float sum = 0.0f;
        for (int kk = 0; kk < KSEL; ++kk)
            sum += w2[o * KSEL + kk] * Vb[cc * C_DIM + perm[C_DIM - KSEL + kk]];
        Tb[idx] = sum;
    }
}

// ---------------------------------------------------------------------------
// Output: out[b] = x[b] + T_b @ gxf[b] + b2. Same 64x64-per-wave blocking.
// ---------------------------------------------------------------------------
__global__ __launch_bounds__(256) void gemm_out_kernel(
    const float* __restrict__ x, const float* __restrict__ T,
    const float* __restrict__ b2, const float* __restrict__ gxf,
    float* __restrict__ out)
{
    const int wave = threadIdx.x >> 5;
    const int lane = threadIdx.x & 31;
    const int half = lane >> 4;
    const int l    = lane & 15;
    const int b     = blockIdx.z;
    const int mBase = blockIdx.y * 64;
    const int nBase = (blockIdx.x * 8 + wave) * 64;

    const float* Tb = T   + (size_t)b * C_DIM * C_DIM;
    const float* gb = gxf + (size_t)b * C_DIM * HW;
    const float* xb = x   + (size_t)b * C_DIM * HW;
    float*       ob = out + (size_t)b * C_DIM * HW;

    v8f acc[4][4] = {};
    for (int k0 = 0; k0 < C_DIM; k0 += 4) {
        const int ka = k0 + 2 * half;
        v2f af[4], bf[4];
#pragma unroll
        for (int i = 0; i < 4; ++i) {
            const float* pa = Tb + (size_t)(mBase + 16 * i + l) * C_DIM + ka;
            af[i].x = pa[0];
            af[i].y = pa[1];
        }
#pragma unroll
        for (int j = 0; j < 4; ++j) {
            const float* pb = gb + (size_t)ka * HW + nBase + 16 * j + l;
            bf[j].x = pb[0];
            bf[j].y = pb[HW];
        }
        if (k0 + 4 < C_DIM) {
            __builtin_prefetch(gb + (size_t)(ka + 4) * HW + nBase + l, 0, 0);
        }
#pragma unroll
        for (int i = 0; i < 4; ++i)
#pragma unroll
            for (int j = 0; j < 4; ++j)
                acc[i][j] = wmma_f32(af[i], bf[j], acc[i][j]);
    }
#pragma unroll
    for (int i = 0; i < 4; ++i) {
#pragma unroll
        for (int v = 0; v < 8; ++v) {
            const int row  = mBase + 16 * i + v + 8 * half;
            const float bias = b2[row];
#pragma unroll
            for (int j = 0; j < 4; ++j) {
                const size_t o = (size_t)row * HW + nBase + 16 * j + l;
                ob[o] = xb[o] + acc[i][j][v] + bias;
            }
        }
    }
}

// ---------------------------------------------------------------------------
extern "C" void kernel_launch(void* const* d_in, const int* in_sizes, int n_in,
                              void* d_out, int out_size, void* d_ws, size_t ws_size,
                              hipStream_t stream)
{
    const float* x  = (const float*)d_in[0];
    const float* w1 = (const float*)d_in[1];
    const float* b1 = (const float*)d_in[2];
    const float* w2 = (const float*)d_in[3];
    const float* b2 = (const float*)d_in[4];
    float* out = (float*)d_out;

    char* ws = (char*)d_ws;
    const size_t GXF_BYTES = (size_t)BATCH * C_DIM * HW * sizeof(float);    // 128 MB
    const size_t MAT_BYTES = (size_t)BATCH * C_DIM * C_DIM * sizeof(float); // 8 MB
    float* gxf  = (float*)(ws);
    float* cmat = (float*)(ws + GXF_BYTES);
    float* V    = (float*)(ws + GXF_BYTES + MAT_BYTES);
    float* T    = (float*)(ws + GXF_BYTES + 2 * MAT_BYTES);

    gemm_gx_kernel <<<dim3(HW / 512, C_DIM / 64, BATCH), 256, 0, stream>>>(x, w1, b1, gxf);
    mean_sub_kernel<<<dim3(BATCH * C_DIM), 256, 0, stream>>>(gxf);
    gemm_cov_kernel<<<dim3(C_DIM / 64, C_DIM / 64, BATCH), 256, 0, stream>>>(gxf, cmat);
    jacobi_kernel  <<<dim3(BATCH), 1024, 0, stream>>>(cmat, w2, V, T);
    gemm_out_kernel<<<dim3(HW / 512, C_DIM / 64, BATCH), 256, 0, stream>>>(x, T, b2, gxf, out);
}